// GCN_48146583388527
// MI455X (gfx1250) — compile-verified
//
#include <hip/hip_runtime.h>
#include <hip/hip_bf16.h>

typedef __attribute__((ext_vector_type(2))) float v2f;
typedef __attribute__((ext_vector_type(8))) float v8f;

#define DH 128  // feature dim (both layers are 128 wide)

// ---------------------------------------------------------------------------
// deg[i] = 1 (self loop weight)
__global__ void GCN_init_deg(float* __restrict__ deg, int n) {
    int i = blockIdx.x * blockDim.x + threadIdx.x;
    if (i < n) deg[i] = 1.0f;
}

// deg[col[e]] += w[e]
__global__ void GCN_accum_deg(const long long* __restrict__ col,
                              const float* __restrict__ w,
                              float* __restrict__ deg, int e) {
    int i = blockIdx.x * blockDim.x + threadIdx.x;
    if (i < e) atomicAdd(&deg[(int)col[i]], w[i]);
}

// in place: deg -> d^-1/2
__global__ void GCN_dinv(float* __restrict__ deg, int n) {
    int i = blockIdx.x * blockDim.x + threadIdx.x;
    if (i < n) {
        float d = deg[i];
        deg[i] = (d > 0.0f) ? rsqrtf(d) : 0.0f;
    }
}

// ---------------------------------------------------------------------------
// Out[n,128] = A[n,128] @ W[128,128] via V_WMMA_F32_16X16X4_F32.
// One block = 16-row strip; 8 waves, wave w owns 16x16 tile at columns w*16.
// A tile staged in LDS (pad 132 to avoid 64-bank conflicts on column reads).
__global__ __launch_bounds__(256) void GCN_gemm_wmma(
    const float* __restrict__ A, const float* __restrict__ W,
    float* __restrict__ Out, int n) {
    __shared__ float sA[16][132];

    const int m0   = blockIdx.x * 16;
    const int tid  = threadIdx.x;
    const int wave = tid >> 5;          // 0..7 -> column tile
    const int lane = tid & 31;          // wave32
    const int half = lane >> 4;         // 0: lanes 0-15, 1: lanes 16-31
    const int l    = lane & 15;
    const int c0   = wave * 16;

    // cooperative load of 16x128 A strip
    for (int i = tid; i < 16 * DH; i += 256) {
        int r = i >> 7, c = i & (DH - 1);
        int row = m0 + r;
        sA[r][c] = (row < n) ? A[(size_t)row * DH + c] : 0.0f;
    }
    __syncthreads();

    // A 16x4 frag: v0 = K {0|2}, v1 = K {1|3} (lanes 0-15 | 16-31)
    // B 4x16 frag: v0 = row K {0|2}, v1 = row K {1|3}, N = lane&15
    const int koff = 2 * half;
    v8f c = {};
#pragma unroll
    for (int kb = 0; kb < DH; kb += 4) {
        v2f a, b;
        a.x = sA[l][kb + koff];
        a.y = sA[l][kb + koff + 1];
        b.x = W[(size_t)(kb + koff) * DH + c0 + l];
        b.y = W[(size_t)(kb + koff + 1) * DH + c0 + l];
        c = __builtin_amdgcn_wmma_f32_16x16x4_f32(
            /*neg_a=*/false, a, /*neg_b=*/false, b,
            /*c_mod=*/(short)0, c, /*reuse_a=*/false, /*reuse_b=*/false);
    }

    // C/D layout: VGPR r = row (r + 8*half), col = lane&15
#pragma unroll
    for (int r = 0; r < 8; ++r) {
        int row = m0 + r + 8 * half;
        if (row < n) Out[(size_t)row * DH + c0 + l] = c[r];
    }
}

// ---------------------------------------------------------------------------
// out[i,:] = dinv[i]^2 * hW[i,:]   (self-loop term; also initializes out)
__global__ void GCN_selfloop_init(const float* __restrict__ hW,
                                  const float* __restrict__ dinv,
                                  float* __restrict__ out, int n) {
    int i = blockIdx.x * blockDim.x + threadIdx.x;
    if (i < n * DH) {
        int node = i >> 7;
        float d = dinv[node];
        out[i] = d * d * hW[i];
    }
}

// out[col[e],:] += (dinv[row]*w*dinv[col]) * hW[row[e],:]
// 32 lanes per edge, float4 gather, scalar f32 atomics for scatter.
__global__ __launch_bounds__(256) void GCN_edge_scatter(
    const long long* __restrict__ rowI, const long long* __restrict__ colI,
    const float* __restrict__ w, const float* __restrict__ dinv,
    const float* __restrict__ hW, float* __restrict__ out, int e) {
    int lane = threadIdx.x & 31;
    int eid  = blockIdx.x * (blockDim.x >> 5) + (threadIdx.x >> 5);
    if (eid >= e) return;
    int r = (int)rowI[eid];
    int c = (int)colI[eid];
    float norm = dinv[r] * w[eid] * dinv[c];
    const float4* src = (const float4*)(hW + (size_t)r * DH);
    float4 v = src[lane];                       // 32 lanes x 4 = 128 floats
    float* dst = out + (size_t)c * DH + lane * 4;
    atomicAdd(dst + 0, norm * v.x);
    atomicAdd(dst + 1, norm * v.y);
    atomicAdd(dst + 2, norm * v.z);
    atomicAdd(dst + 3, norm * v.w);
}

// h[i,f] = (relu ? max(h+b,0) : h+b)
__global__ void GCN_bias_act(float* __restrict__ h, const float* __restrict__ b,
                             int n, int do_relu) {
    int i = blockIdx.x * blockDim.x + threadIdx.x;
    if (i < n * DH) {
        float v = h[i] + b[i & (DH - 1)];
        h[i] = do_relu ? fmaxf(v, 0.0f) : v;
    }
}

// ---------------------------------------------------------------------------
extern "C" void kernel_launch(void* const* d_in, const int* in_sizes, int n_in,
                              void* d_out, int out_size, void* d_ws, size_t ws_size,
                              hipStream_t stream) {
    const float*     x   = (const float*)d_in[0];
    const long long* ei  = (const long long*)d_in[1];  // int64, shape (2,E) flat
    const float*     ew  = (const float*)d_in[2];
    const float*     W1  = (const float*)d_in[3];
    const float*     b1  = (const float*)d_in[4];
    const float*     W2  = (const float*)d_in[5];
    const float*     b2  = (const float*)d_in[6];

    const int N = in_sizes[0] / DH;
    const int E = in_sizes[2];
    const long long* rowI = ei;       // sources
    const long long* colI = ei + E;   // targets

    float* ws   = (float*)d_ws;
    float* dinv = ws;                 // N floats
    float* hW   = ws + N;             // N*128 (GEMM products, reused)
    float* h    = hW + (size_t)N * DH; // N*128 (aggregated hidden)

    float* out = (float*)d_out;

    const int T = 256;
    const int gN   = (N + T - 1) / T;
    const int gE   = (E + T - 1) / T;
    const int gNF  = (N * DH + T - 1) / T;
    const int gEw  = (E + (T / 32) - 1) / (T / 32);  // 8 edges per block
    const int gMM  = (N + 15) / 16;

    // normalization
    GCN_init_deg  <<<gN,  T, 0, stream>>>(dinv, N);
    GCN_accum_deg <<<gE,  T, 0, stream>>>(colI, ew, dinv, E);
    GCN_dinv      <<<gN,  T, 0, stream>>>(dinv, N);

    // layer 1: hW = x @ W1 ; h = scatter(norm * hW) + b1 ; relu
    GCN_gemm_wmma    <<<gMM, T, 0, stream>>>(x, W1, hW, N);
    GCN_selfloop_init<<<gNF, T, 0, stream>>>(hW, dinv, h, N);
    GCN_edge_scatter <<<gEw, T, 0, stream>>>(rowI, colI, ew, dinv, hW, h, E);
    GCN_bias_act     <<<gNF, T, 0, stream>>>(h, b1, N, 1);

    // layer 2: hW = h @ W2 ; out = scatter(norm * hW) + b2
    GCN_gemm_wmma    <<<gMM, T, 0, stream>>>(h, W2, hW, N);
    GCN_selfloop_init<<<gNF, T, 0, stream>>>(hW, dinv, out, N);
    GCN_edge_scatter <<<gEw, T, 0, stream>>>(rowI, colI, ew, dinv, hW, out, E);
    GCN_bias_act     <<<gNF, T, 0, stream>>>(out, b2, N, 0);
}